// APNLayer_38259568673551
// MI455X (gfx1250) — compile-verified
//
#include <hip/hip_runtime.h>
#include <hip/hip_bf16.h>

#define B_ 4
#define L_ 2048
#define D_ 512
#define CHUNK_ 64
#define NCH (L_ / CHUNK_)
#define VW 32           // value-dim slice per workgroup in the scan kernel
#define KSTR 516        // Kt row stride (16B-aligned rows, conflict-free)
#define EPS_ 1e-5f

typedef float v2f __attribute__((ext_vector_type(2)));
typedef float v8f __attribute__((ext_vector_type(8)));

__device__ __forceinline__ v8f wmma_f32(v2f a, v2f b, v8f c) {
  // D = A(16x4 f32) * B(4x16 f32) + C(16x16 f32)
  return __builtin_amdgcn_wmma_f32_16x16x4_f32(false, a, false, b, (short)0, c,
                                               false, false);
}

// Async HBM -> LDS DMA (ASYNCcnt-tracked).  vDst = per-lane LDS byte offset.
__device__ __forceinline__ void async_b128(const float* g, unsigned lds_byte_off) {
  asm volatile("global_load_async_to_lds_b128 %0, %1, off"
               :
               : "v"(lds_byte_off), "v"(g)
               : "memory");
}
__device__ __forceinline__ void wait_async_lds() {
  asm volatile("s_wait_asynccnt 0x0" ::: "memory");
}

// ---------------------------------------------------------------- kernel 1
__global__ void tanh_k(const float* __restrict__ x, float* __restrict__ xa, int n) {
  int i = blockIdx.x * blockDim.x + threadIdx.x;
  if (i < n) xa[i] = tanhf(x[i]);
}

// ---------------------------------------------------------------- kernel 2
// static_out[m, e] = sum_d xa[m, d] * W[e, d]   (M=8192, N=512, K=512)
__global__ __launch_bounds__(256) void gemm_xw(const float* __restrict__ xa,
                                               const float* __restrict__ Wm,
                                               float* __restrict__ out) {
  __shared__ float At[64 * 33];   // A panel [64 rows][32 k] padded
  __shared__ float Bt[32 * 65];   // B panel [32 k][64 n] padded (B[k][n] = W[n][k])
  const int tid = threadIdx.x, lane = tid & 31, w = tid >> 5;
  const int m0 = blockIdx.x * 64, n0 = blockIdx.y * 64;
  const int wm = w & 3, wn = w >> 2;           // 4 x 2 wave grid
  const int nn = lane & 15;
  const int kbo = (lane >> 4) << 1;
  const int rh = (lane >> 4) << 3;

  v8f c0 = {}, c1 = {};
  for (int kp = 0; kp < D_; kp += 32) {
    for (int idx = tid; idx < 64 * 32; idx += 256) {
      int r = idx >> 5, c = idx & 31;
      At[r * 33 + c] = xa[(size_t)(m0 + r) * D_ + kp + c];
    }
    for (int idx = tid; idx < 64 * 32; idx += 256) {
      int k = idx & 31, n = idx >> 5;
      Bt[k * 65 + n] = Wm[(size_t)(n0 + n) * D_ + kp + k];
    }
    __syncthreads();
#pragma unroll
    for (int kk = 0; kk < 32; kk += 4) {
      int kb = kk + kbo;
      v2f a  = { At[(wm * 16 + nn) * 33 + kb], At[(wm * 16 + nn) * 33 + kb + 1] };
      v2f b0 = { Bt[kb * 65 + wn * 32 + nn], Bt[(kb + 1) * 65 + wn * 32 + nn] };
      v2f b1 = { Bt[kb * 65 + wn * 32 + 16 + nn], Bt[(kb + 1) * 65 + wn * 32 + 16 + nn] };
      c0 = wmma_f32(a, b0, c0);
      c1 = wmma_f32(a, b1, c1);
    }
    __syncthreads();
  }
#pragma unroll
  for (int i = 0; i < 8; ++i) {
    int r = m0 + wm * 16 + i + rh;
    out[(size_t)r * D_ + n0 + wn * 32 + nn]      = c0[i];
    out[(size_t)r * D_ + n0 + wn * 32 + 16 + nn] = c1[i];
  }
}

// ---------------------------------------------------------------- kernel 3
// Chunked gated delta-rule scan for one (batch, value-slice), q == k.
// Dynamic-LDS layout (byte offsets are compile-time constants; no static LDS,
// so the dynamic region starts at LDS offset 0 -> usable by async DMA):
#define S_OFF   0u
#define KT_OFF  (512u * VW * 4u)
#define UT_OFF  (KT_OFF + 64u * KSTR * 4u)
#define AT_OFF  (UT_OFF + 64u * 33u * 4u)
#define GS_OFF  (AT_OFF + 64u * 33u * 4u)
#define LP_OFF  (GS_OFF + 64u * 65u * 4u)
#define SMEM_FLOATS (512 * VW + 64 * KSTR + 2 * 64 * 33 + 64 * 65 + 65)

__global__ __launch_bounds__(256) void gdn_scan(const float* __restrict__ xa,
                                                const float* __restrict__ vstat,
                                                const float* __restrict__ eta_p,
                                                const float* __restrict__ laml_p,
                                                float* __restrict__ y) {
  extern __shared__ float smem[];
  float* S      = smem;                 // [512][VW]
  float* Kt     = S + 512 * VW;         // [64][KSTR]
  float* Ut     = Kt + 64 * KSTR;       // [64][33]
  float* At     = Ut + 64 * 33;         // [64][33]
  float* GS     = At + 64 * 33;         // [64][65]
  float* lampow = GS + 64 * 65;         // [65]

  const int tid = threadIdx.x, lane = tid & 31, w = tid >> 5;
  const int b  = blockIdx.x >> 4;             // 4 batches x 16 slices
  const int v0 = (blockIdx.x & 15) * VW;
  const float beta = eta_p[0];
  const float lam  = 1.f / (1.f + __expf(-laml_p[0]));

  for (int i = tid; i < 512 * VW; i += 256) S[i] = 0.f;
  if (tid < 65) lampow[tid] = powf(lam, (float)tid);

  const size_t bbase = (size_t)b * L_ * D_;
  // async-stage chunk 0's K (=Q) block
  for (int i = tid; i < CHUNK_ * D_ / 4; i += 256) {
    int t = i >> 7, d4 = (i & 127) << 2;
    async_b128(xa + bbase + (size_t)t * D_ + d4,
               KT_OFF + (unsigned)((t * KSTR + d4) * 4));
  }

  const int nn  = lane & 15;            // N index inside a 16-wide tile
  const int kbo = (lane >> 4) << 1;     // K sub-offset of the f32 A/B fragment
  const int rh  = (lane >> 4) << 3;     // row-half offset of C/D fragment
  const int tm = w & 3, tn = w >> 2;    // wave tile coords (4 x 2)
  const int rowA = tm * 16 + nn;

  for (int ch = 0; ch < NCH; ++ch) {
    const size_t xbase = bbase + (size_t)ch * CHUNK_ * D_;
    wait_async_lds();
    __syncthreads();                    // Kt ready; S/At/GS/Ut consistent
    const float lamC = lampow[64];

    // ---- prefetch next chunk (K + V slice) into cache while we compute
    if (ch + 1 < NCH) {
      const float* nk = xa + xbase + (size_t)CHUNK_ * D_;
      for (int i = tid; i < CHUNK_ * D_ / 32; i += 256)
        __builtin_prefetch(nk + i * 32, 0, 3);
      if (tid < CHUNK_)
        __builtin_prefetch(vstat + xbase + (size_t)(CHUNK_ + tid) * D_ + v0, 0, 3);
    }

    // ---- A = K@S [64x32], GS = mask/decay(K@K^T) [64x64]
    v8f aA = {}, aG0 = {}, aG1 = {};
    for (int k0 = 0; k0 < D_; k0 += 4) {
      int kb = k0 + kbo;
      v2f a  = { Kt[rowA * KSTR + kb], Kt[rowA * KSTR + kb + 1] };
      v2f bS = { S[kb * VW + tn * 16 + nn], S[(kb + 1) * VW + tn * 16 + nn] };
      aA = wmma_f32(a, bS, aA);
      v2f b0 = { Kt[((2 * tn) * 16 + nn) * KSTR + kb],
                 Kt[((2 * tn) * 16 + nn) * KSTR + kb + 1] };
      v2f b1 = { Kt[((2 * tn + 1) * 16 + nn) * KSTR + kb],
                 Kt[((2 * tn + 1) * 16 + nn) * KSTR + kb + 1] };
      aG0 = wmma_f32(a, b0, aG0);
      aG1 = wmma_f32(a, b1, aG1);
    }
#pragma unroll
    for (int i = 0; i < 8; ++i) {
      int t = tm * 16 + i + rh;
      At[t * 33 + tn * 16 + nn] = aA[i];
      int s0 = (2 * tn) * 16 + nn, s1 = s0 + 16;
      GS[t * 65 + s0] = (s0 <= t) ? lampow[t - s0] * aG0[i] : 0.f;
      GS[t * 65 + s1] = (s1 <= t) ? lampow[t - s1] * aG1[i] : 0.f;
    }
    __syncthreads();

    // ---- RHS: U <- beta*(V - lam^{t+1} A)   (V straight from global/L2)
    for (int i = tid; i < CHUNK_ * VW; i += 256) {
      int t = i >> 5, j = i & 31;
      float v = vstat[xbase + (size_t)t * D_ + v0 + j];
      Ut[t * 33 + j] = beta * (v - lampow[t + 1] * At[t * 33 + j]);
    }
    __syncthreads();

    // ---- blocked forward substitution: (I + beta*strictlow(GS)) U = R
    // (e1) rows 0..31; wave w owns value-columns [4w, 4w+4)
    for (int t = 1; t < 32; ++t) {
#pragma unroll
      for (int jj = 0; jj < 4; ++jj) {
        int j = w * 4 + jj;
        float p = (lane < t) ? GS[t * 65 + lane] * Ut[lane * 33 + j] : 0.f;
#pragma unroll
        for (int off = 16; off > 0; off >>= 1) p += __shfl_xor(p, off, 32);
        if (lane == 0) Ut[t * 33 + j] -= beta * p;
      }
    }
    __syncthreads();
    // (e2) U_hi -= beta * GS[32:64,0:32] @ U_lo   (WMMA, -beta folded into A)
    if (w < 4) {
      const int tm2 = w & 1, tn2 = w >> 1;
      const float nbeta = -beta;
      v8f acc;
#pragma unroll
      for (int i = 0; i < 8; ++i)
        acc[i] = Ut[(32 + tm2 * 16 + i + rh) * 33 + tn2 * 16 + nn];
#pragma unroll
      for (int k0 = 0; k0 < 32; k0 += 4) {
        int kb = k0 + kbo;
        v2f a  = { nbeta * GS[(32 + tm2 * 16 + nn) * 65 + kb],
                   nbeta * GS[(32 + tm2 * 16 + nn) * 65 + kb + 1] };
        v2f bU = { Ut[kb * 33 + tn2 * 16 + nn], Ut[(kb + 1) * 33 + tn2 * 16 + nn] };
        acc = wmma_f32(a, bU, acc);
      }
#pragma unroll
      for (int i = 0; i < 8; ++i)
        Ut[(32 + tm2 * 16 + i + rh) * 33 + tn2 * 16 + nn] = acc[i];
    }
    __syncthreads();
    // (e3) rows 33..63, remaining strictly-lower terms s in [32, t)
    for (int t = 33; t < CHUNK_; ++t) {
#pragma unroll
      for (int jj = 0; jj < 4; ++jj) {
        int j = w * 4 + jj;
        int s = 32 + lane;
        float p = (s < t) ? GS[t * 65 + s] * Ut[s * 33 + j] : 0.f;
#pragma unroll
        for (int off = 16; off > 0; off >>= 1) p += __shfl_xor(p, off, 32);
        if (lane == 0) Ut[t * 33 + j] -= beta * p;
      }
    }
    __syncthreads();

    // ---- state carry FIRST (last Kt consumer): S = lam^64 S + (lam^{63-s}k_s)^T U
    for (int r = 0; r < 4; ++r) {
      int mt = w + r * 8;
      for (int t2 = 0; t2 < 2; ++t2) {
        v8f acc;
#pragma unroll
        for (int i = 0; i < 8; ++i)
          acc[i] = lamC * S[(mt * 16 + i + rh) * VW + t2 * 16 + nn];
        for (int k0 = 0; k0 < CHUNK_; k0 += 4) {
          int kb = k0 + kbo;
          v2f a  = { lampow[63 - kb] * Kt[kb * KSTR + mt * 16 + nn],
                     lampow[62 - kb] * Kt[(kb + 1) * KSTR + mt * 16 + nn] };
          v2f bU = { Ut[kb * 33 + t2 * 16 + nn], Ut[(kb + 1) * 33 + t2 * 16 + nn] };
          acc = wmma_f32(a, bU, acc);
        }
#pragma unroll
        for (int i = 0; i < 8; ++i)
          S[(mt * 16 + i + rh) * VW + t2 * 16 + nn] = acc[i];
      }
    }
    __syncthreads();                     // all waves done reading Kt

    // ---- kick off next chunk's async K copy; it overlaps the O-matmul below
    if (ch + 1 < NCH) {
      const float* nk = xa + xbase + (size_t)CHUNK_ * D_;
      for (int i = tid; i < CHUNK_ * D_ / 4; i += 256) {
        int t = i >> 7, d4 = (i & 127) << 2;
        async_b128(nk + (size_t)t * D_ + d4,
                   KT_OFF + (unsigned)((t * KSTR + d4) * 4));
      }
    }

    // ---- O = GS @ U ; y = V + lam^{t+1} A + O
    v8f aO = {};
    for (int k0 = 0; k0 < CHUNK_; k0 += 4) {
      int kb = k0 + kbo;
      v2f a  = { GS[rowA * 65 + kb], GS[rowA * 65 + kb + 1] };
      v2f bU = { Ut[kb * 33 + tn * 16 + nn], Ut[(kb + 1) * 33 + tn * 16 + nn] };
      aO = wmma_f32(a, bU, aO);
    }
#pragma unroll
    for (int i = 0; i < 8; ++i) {
      int t = tm * 16 + i + rh;
      int j = tn * 16 + nn;
      float v = vstat[xbase + (size_t)t * D_ + v0 + j];
      y[xbase + (size_t)t * D_ + v0 + j] =
          v + lampow[t + 1] * At[t * 33 + j] + aO[i];
    }
    // loop-top wait_async + barrier orders these reads vs next chunk's DMA
  }
}

// ---------------------------------------------------------------- kernel 4
__global__ __launch_bounds__(256) void layernorm_k(float* __restrict__ y,
                                                   const float* __restrict__ gam,
                                                   const float* __restrict__ bet) {
  __shared__ float rs[256];
  __shared__ float rq[256];
  const int row = blockIdx.x, tid = threadIdx.x;
  float v0 = y[(size_t)row * D_ + tid];
  float v1 = y[(size_t)row * D_ + tid + 256];
  rs[tid] = v0 + v1;
  rq[tid] = v0 * v0 + v1 * v1;
  __syncthreads();
  for (int s = 128; s > 0; s >>= 1) {
    if (tid < s) { rs[tid] += rs[tid + s]; rq[tid] += rq[tid + s]; }
    __syncthreads();
  }
  float mean = rs[0] * (1.f / D_);
  float var  = rq[0] * (1.f / D_) - mean * mean;
  float inv  = rsqrtf(var + EPS_);
  y[(size_t)row * D_ + tid]       = (v0 - mean) * inv * gam[tid] + bet[tid];
  y[(size_t)row * D_ + tid + 256] = (v1 - mean) * inv * gam[tid + 256] + bet[tid + 256];
}

// ---------------------------------------------------------------- launch
extern "C" void kernel_launch(void* const* d_in, const int* in_sizes, int n_in,
                              void* d_out, int out_size, void* d_ws, size_t ws_size,
                              hipStream_t stream) {
  const float* x    = (const float*)d_in[0];
  const float* W    = (const float*)d_in[1];
  const float* eta  = (const float*)d_in[2];
  const float* laml = (const float*)d_in[3];
  const float* g    = (const float*)d_in[4];
  const float* bb   = (const float*)d_in[5];
  float* out = (float*)d_out;

  float* xa    = (float*)d_ws;                       // [B,L,D] tanh(x)
  float* vstat = xa + (size_t)B_ * L_ * D_;          // [B,L,D] static_out

  const int n = B_ * L_ * D_;
  tanh_k<<<(n + 255) / 256, 256, 0, stream>>>(x, xa, n);

  dim3 g2(B_ * L_ / 64, D_ / 64);
  gemm_xw<<<g2, 256, 0, stream>>>(xa, W, vstat);

  size_t smem = (size_t)SMEM_FLOATS * sizeof(float);
  gdn_scan<<<B_ * (D_ / VW), 256, smem, stream>>>(xa, vstat, eta, laml, out);

  layernorm_k<<<B_ * L_, 256, 0, stream>>>(out, g, bb);
}